// ESN_3487513444948
// MI455X (gfx1250) — compile-verified
//
#include <hip/hip_runtime.h>
#include <hip/hip_bf16.h>
#include <math.h>

#define T_STEPS 8192
#define NU      128
#define NX      2048
#define NY      128
#define ALPHA_F 0.95f

typedef float v2f __attribute__((ext_vector_type(2)));
typedef float v8f __attribute__((ext_vector_type(8)));

// ---------------------------------------------------------------------------
// Register-blocked fp32 WMMA GEMM core: one wave32 computes a 32x32 tile of
// C[M,N] = A[M,K] * B[N,K]^T  (both operands row-major with row stride LDK).
// 2x2 grid of 16x16 WMMA accumulators; each fragment load feeds two WMMAs
// -> ~8 FLOP/byte from L2 instead of 4.
// ---------------------------------------------------------------------------
template <int LDK, int LDC>
__device__ __forceinline__ void wmma_tile_32x32(const float* __restrict__ A,
                                                const float* __restrict__ Bm,
                                                float* __restrict__ C,
                                                int m0, int n0) {
  const int lane = threadIdx.x & 31;
  const int n  = lane & 15;                 // M index for A-frag, N for B/C/D
  const int kb = (lane >> 4) << 1;          // K sub-offset {0,2}

  const float* a0 = A  + (size_t)(m0 + n) * LDK;
  const float* a1 = a0 + (size_t)16 * LDK;
  const float* b0 = Bm + (size_t)(n0 + n) * LDK;   // B-role[k][n] = Bm[n][k]
  const float* b1 = b0 + (size_t)16 * LDK;

  v8f c00 = {}, c01 = {}, c10 = {}, c11 = {};
#pragma unroll 2
  for (int k0 = 0; k0 < LDK; k0 += 4) {
    float2 fa0 = *(const float2*)(a0 + k0 + kb);
    float2 fa1 = *(const float2*)(a1 + k0 + kb);
    float2 fb0 = *(const float2*)(b0 + k0 + kb);
    float2 fb1 = *(const float2*)(b1 + k0 + kb);
    v2f va0; va0.x = fa0.x; va0.y = fa0.y;
    v2f va1; va1.x = fa1.x; va1.y = fa1.y;
    v2f vb0; vb0.x = fb0.x; vb0.y = fb0.y;
    v2f vb1; vb1.x = fb1.x; vb1.y = fb1.y;
    c00 = __builtin_amdgcn_wmma_f32_16x16x4_f32(false, va0, false, vb0,
                                                (short)0, c00, false, false);
    c01 = __builtin_amdgcn_wmma_f32_16x16x4_f32(false, va0, false, vb1,
                                                (short)0, c01, false, false);
    c10 = __builtin_amdgcn_wmma_f32_16x16x4_f32(false, va1, false, vb0,
                                                (short)0, c10, false, false);
    c11 = __builtin_amdgcn_wmma_f32_16x16x4_f32(false, va1, false, vb1,
                                                (short)0, c11, false, false);
  }
  const int mhi = (lane >> 4) * 8;
#pragma unroll
  for (int r = 0; r < 8; ++r) {
    C[(size_t)(m0 +      r + mhi) * LDC + (n0      + n)] = c00[r];
    C[(size_t)(m0 +      r + mhi) * LDC + (n0 + 16 + n)] = c01[r];
    C[(size_t)(m0 + 16 + r + mhi) * LDC + (n0      + n)] = c10[r];
    C[(size_t)(m0 + 16 + r + mhi) * LDC + (n0 + 16 + n)] = c11[r];
  }
}

// Kernel 1: B[t,i] = sum_k UT[t,k] * W_in[i,k]  -> staged in X region
__global__ __launch_bounds__(256)
void esn_inproj(const float* __restrict__ UT, const float* __restrict__ Win,
                float* __restrict__ B) {
  const int wid  = blockIdx.x * (blockDim.x >> 5) + (threadIdx.x >> 5);
  const int tilesN = NX / 32;                 // 64
  const int tM = wid / tilesN;                // [0,256)
  const int tN = wid % tilesN;                // [0,64)
  wmma_tile_32x32<NU, NX>(UT, Win, B, tM * 32, tN * 32);
}

// Kernel 3: Y[t,j] = sum_i X[t,i] * W_out[j,i]
__global__ __launch_bounds__(256)
void esn_outproj(const float* __restrict__ X, const float* __restrict__ Wout,
                 float* __restrict__ Y) {
  const int wid  = blockIdx.x * (blockDim.x >> 5) + (threadIdx.x >> 5);
  const int tilesN = NY / 32;                 // 4
  const int tM = wid / tilesN;                // [0,256)
  const int tN = wid % tilesN;                // [0,4)
  wmma_tile_32x32<NX, NY>(X, Wout, Y, tM * 32, tN * 32);
}

// ---------------------------------------------------------------------------
// Kernel 2: persistent sequential scan.
// 16 workgroups x 1024 threads. Each WG owns 128 rows of W, compressed to
// ELL (5% density -> ~102 nnz/row, capacity 224) resident in LDS (~182 KB).
// Per step: 8 threads/row ELL-dot against LDS-resident x, shfl reduce,
// leaky-tanh update, global barrier, refill x from the just-written X row.
// ---------------------------------------------------------------------------
#define NWG       16
#define WGSIZE    1024
#define ROWS_WG   (NX / NWG)        // 128
#define ELLW      224               // nnz capacity per row (mean 102, +12 sigma)
#define ELLSTR    232               // padded LDS stride (bank walk)
#define SLOTS_THR (ELLW / 8)        // 28 slots per thread

#define SMEM_BYTES (ROWS_WG * ELLSTR * 4 /*vals*/ + \
                    ROWS_WG * ELLSTR * 2 /*cols*/ + NX * 4 /*x*/)

__device__ __forceinline__ void grid_barrier(int* cnt, int* gen) {
  __syncthreads();
  if (threadIdx.x == 0) {
    __threadfence();
    int g = __hip_atomic_load(gen, __ATOMIC_RELAXED, __HIP_MEMORY_SCOPE_AGENT);
    int arrived = __hip_atomic_fetch_add(cnt, 1, __ATOMIC_ACQ_REL,
                                         __HIP_MEMORY_SCOPE_AGENT) + 1;
    if (arrived == NWG) {
      __hip_atomic_store(cnt, 0, __ATOMIC_RELAXED, __HIP_MEMORY_SCOPE_AGENT);
      __hip_atomic_fetch_add(gen, 1, __ATOMIC_RELEASE, __HIP_MEMORY_SCOPE_AGENT);
    } else {
      while (__hip_atomic_load(gen, __ATOMIC_ACQUIRE,
                               __HIP_MEMORY_SCOPE_AGENT) == g) { }
    }
  }
  __syncthreads();
}

__global__ __launch_bounds__(WGSIZE)
void esn_scan(const float* __restrict__ W, float* __restrict__ X,
              int* __restrict__ bar) {
  extern __shared__ char smem[];
  float*          ell_val = (float*)smem;
  unsigned short* ell_col = (unsigned short*)(ell_val + ROWS_WG * ELLSTR);
  float*          x_lds   = (float*)(ell_col + ROWS_WG * ELLSTR);

  const int wave = threadIdx.x >> 5;
  const int lane = threadIdx.x & 31;
  const int wg_base = blockIdx.x * ROWS_WG;

  // ---- one-time build: dense 5%-sparse W rows -> LDS ELL ----
  for (int rr = wave * 4; rr < wave * 4 + 4; ++rr) {      // 32 waves * 4 rows
    const float* wrow = W + (size_t)(wg_base + rr) * NX;
    int cnt = 0;
    for (int c0 = 0; c0 < NX; c0 += 32) {
      __builtin_prefetch(wrow + c0 + 256, 0, 1);          // global_prefetch_b8
      float v = wrow[c0 + lane];
      unsigned long long m = __ballot(v != 0.0f);         // wave32: low 32 bits
      int prefix = __popcll(m & ((1ull << lane) - 1ull));
      if (v != 0.0f) {
        int slot = cnt + prefix;
        if (slot < ELLW) {
          ell_val[rr * ELLSTR + slot] = v;
          ell_col[rr * ELLSTR + slot] = (unsigned short)(c0 + lane);
        }
      }
      cnt += (int)__popcll(m);
    }
    for (int s = cnt + lane; s < ELLW; s += 32) {         // pad
      ell_val[rr * ELLSTR + s] = 0.0f;
      ell_col[rr * ELLSTR + s] = 0;
    }
  }
  for (int i = threadIdx.x; i < NX; i += WGSIZE) x_lds[i] = 0.0f;  // x_{-1}=0
  __syncthreads();

  // ---- sequential recurrence ----
  const int r = threadIdx.x >> 3;         // local row  [0,128)
  const int s = threadIdx.x & 7;          // slot group [0,8)
  const int grow = wg_base + r;
  const float* vp = ell_val + r * ELLSTR + s * SLOTS_THR;
  const unsigned short* cp = ell_col + r * ELLSTR + s * SLOTS_THR;

  for (int t = 0; t < T_STEPS; ++t) {
    float partial = 0.0f;
#pragma unroll
    for (int j = 0; j < SLOTS_THR; ++j)
      partial += vp[j] * x_lds[cp[j]];
    partial += __shfl_xor(partial, 1, 8);
    partial += __shfl_xor(partial, 2, 8);
    partial += __shfl_xor(partial, 4, 8);
    if (s == 0) {
      float b  = X[(size_t)t * NX + grow];                // staged B[t,row]
      float xn = (1.0f - ALPHA_F) * x_lds[grow] + ALPHA_F * tanhf(b + partial);
      X[(size_t)t * NX + grow] = xn;                      // overwrite in place
    }
    grid_barrier(bar, bar + 1);
    for (int i = threadIdx.x; i < NX; i += WGSIZE)        // broadcast x_t
      x_lds[i] = X[(size_t)t * NX + i];
    __syncthreads();
  }
}

__global__ void esn_bar_init(int* bar) { bar[0] = 0; bar[1] = 0; }

// ---------------------------------------------------------------------------
extern "C" void kernel_launch(void* const* d_in, const int* in_sizes, int n_in,
                              void* d_out, int out_size, void* d_ws, size_t ws_size,
                              hipStream_t stream) {
  (void)in_sizes; (void)n_in; (void)out_size; (void)ws_size;
  const float* UT   = (const float*)d_in[0];   // [T, NU]
  const float* Win  = (const float*)d_in[1];   // [NX, NU]
  const float* W    = (const float*)d_in[2];   // [NX, NX]
  const float* Wout = (const float*)d_in[3];   // [NY, NX]

  float* Y = (float*)d_out;                    // [T, NY]
  float* X = Y + (size_t)T_STEPS * NY;         // [T, NX] (also stages B)
  int*   bar = (int*)d_ws;                     // 2 ints of barrier state

  // allow >64KB dynamic LDS for the scan kernel (idempotent, deterministic)
  hipFuncSetAttribute((const void*)esn_scan,
                      hipFuncAttributeMaxDynamicSharedMemorySize, SMEM_BYTES);

  esn_bar_init<<<1, 1, 0, stream>>>(bar);

  // B = UT @ Win^T  -> staged into X region (32x32 tiles, 8 waves/block)
  {
    const int tiles = (T_STEPS / 32) * (NX / 32);         // 16384
    esn_inproj<<<tiles / 8, 256, 0, stream>>>(UT, Win, X);
  }

  // sequential leaky-tanh scan with LDS-resident sparse W
  esn_scan<<<NWG, WGSIZE, SMEM_BYTES, stream>>>(W, X, bar);

  // Y = X @ Wout^T
  {
    const int tiles = (T_STEPS / 32) * (NY / 32);         // 1024
    esn_outproj<<<tiles / 8, 256, 0, stream>>>(X, Wout, Y);
  }
}